// DisplaceChannel_37099927503239
// MI455X (gfx1250) — compile-verified
//
#include <hip/hip_runtime.h>
#include <cstdint>

// DisplaceChannel on MI455X: pure bandwidth problem (~537 MB moved, ~23 us
// floor @ 23.3 TB/s, ~7 flop/elem). Per-group constant offset => constant
// integer shift (dy,dx) + constant bilinear weights. Strategy: async b128
// global->LDS staging of unshifted rows (aligned, 16B/lane), zero-padded LDS
// margins so the shifted bilinear combine needs no bounds checks, and
// non-temporal 128-bit output stores.

typedef float v4f __attribute__((ext_vector_type(4)));

constexpr int Bv = 16;
constexpr int Cv = 256;
constexpr int Hv = 128;
constexpr int Wv = 128;
constexpr int BIND = 8;            // channels per offset group (256/32)
constexpr int TILE_H = 32;         // output rows per block
constexpr int WIN_R = TILE_H + 1;  // 33 staged rows
constexpr int PAD = 16;            // zero margin (covers |dx|<=13 + 4-tap reach)
constexpr int WS = PAD + Wv + PAD; // 160 floats; row base stays 16B-aligned
constexpr int NTHREADS = 256;

__global__ __launch_bounds__(NTHREADS)
void displace_kernel(const float* __restrict__ inp,
                     const float* __restrict__ offs,
                     float* __restrict__ out)
{
    __shared__ float win[WIN_R * WS];   // 33*160*4 = 21120 bytes

    const int tile = blockIdx.x;
    const int bc   = tile >> 2;                 // image index b*C + c
    const int y0   = (tile & 3) * TILE_H;       // first output row of tile
    const int c    = bc & (Cv - 1);
    const int g    = c / BIND;

    // Absolute offsets; integer shift + constant bilinear weights.
    const float ox = offs[2 * g + 0] * 128.0f;
    const float oy = offs[2 * g + 1] * 128.0f;
    const int   dx = (int)floorf(ox);
    const int   dy = (int)floorf(oy);
    const float fx = ox - (float)dx;
    const float fy = oy - (float)dy;
    const float w00 = (1.0f - fy) * (1.0f - fx);
    const float w01 = (1.0f - fy) * fx;
    const float w10 = fy * (1.0f - fx);
    const float w11 = fy * fx;

    const float* src = inp + (size_t)bc * (Hv * Wv);
    const int tid = threadIdx.x;
    const v4f zero = {0.f, 0.f, 0.f, 0.f};

    // ---- Zero the left/right pad margins (8 x 16B chunks per row) ----
    for (int i = tid; i < WIN_R * 8; i += NTHREADS) {
        const int r = i >> 3;
        const int k = i & 7;                       // 0..3 left, 4..7 right
        const int col = (k < 4) ? (k * 4) : (PAD + Wv + (k - 4) * 4);
        *(v4f*)&win[r * WS + col] = zero;
    }

    // ---- Stage rows y0+dy .. y0+dy+32 into LDS interior (async b128) ----
    // win[r][PAD+col] = I[y0+dy+r][col]; fully OOB rows are zero-filled.
    for (int i = tid; i < WIN_R * (Wv / 4); i += NTHREADS) {
        const int r = i >> 5;                      // staged row
        const int l = i & 31;                      // 16B chunk within row
        const int gy = y0 + dy + r;
        float* lp = &win[r * WS + PAD + l * 4];
        if ((unsigned)gy < (unsigned)Hv) {
            // Low 32 bits of the flat shared-aperture pointer == LDS byte
            // offset relative to the wave's LDS_BASE (ISA 10.2 aperture map).
            const unsigned lds_addr = (unsigned)(uintptr_t)lp;
            const unsigned long long gva =
                (unsigned long long)(uintptr_t)(src + gy * Wv + l * 4);
            asm volatile("global_load_async_to_lds_b128 %0, %1, off th:TH_LOAD_NT"
                         :: "v"(lds_addr), "v"(gva)
                         : "memory");
        } else {
            *(v4f*)lp = zero;
        }
    }
    // Drain this wave's async copies, then workgroup barrier.
    asm volatile("s_wait_asynccnt 0" ::: "memory");
    __syncthreads();

    // ---- Shifted bilinear 2x2 combine from LDS; 128-bit NT stores ----
    v4f* dst = (v4f*)(out + (size_t)bc * (Hv * Wv) + (size_t)y0 * Wv);
    for (int i = tid; i < (TILE_H * Wv) / 4; i += NTHREADS) {
        const int ly = i >> 5;             // row in tile   (W/4 = 32 quads)
        const int x4 = (i & 31) << 2;      // first column of the quad
        const float* r0 = &win[ly * WS + PAD + x4 + dx];
        const float* r1 = r0 + WS;
        const float a0 = r0[0], a1 = r0[1], a2 = r0[2], a3 = r0[3], a4 = r0[4];
        const float b0 = r1[0], b1 = r1[1], b2 = r1[2], b3 = r1[3], b4 = r1[4];
        v4f v;
        v.x = w00 * a0 + w01 * a1 + w10 * b0 + w11 * b1;
        v.y = w00 * a1 + w01 * a2 + w10 * b1 + w11 * b2;
        v.z = w00 * a2 + w01 * a3 + w10 * b2 + w11 * b3;
        v.w = w00 * a3 + w01 * a4 + w10 * b3 + w11 * b4;
        __builtin_nontemporal_store(v, dst + i);
    }
}

extern "C" void kernel_launch(void* const* d_in, const int* in_sizes, int n_in,
                              void* d_out, int out_size, void* d_ws, size_t ws_size,
                              hipStream_t stream)
{
    const float* inp  = (const float*)d_in[0];   // [16,256,128,128] fp32
    const float* offs = (const float*)d_in[1];   // [32,2] fp32
    float* out = (float*)d_out;                  // [16,256,128,128] fp32

    const int nblocks = Bv * Cv * (Hv / TILE_H); // 16384 tiles
    displace_kernel<<<nblocks, NTHREADS, 0, stream>>>(inp, offs, out);
}